// Data_32985348833820
// MI455X (gfx1250) — compile-verified
//
#include <hip/hip_runtime.h>
#include <stdint.h>

#define B2      64
#define BATCH   32
#define NAP     256
#define NUD     2560
#define EXW     100.0f

// ---------------------------------------------------------------------------
// CDNA5 async global->LDS DMA helpers (gfx1250). Inline asm per bridge doc:
// portable across ROCm7.2 / amdgpu-toolchain (bypasses builtin arity skew).
// Generic pointers to __shared__ have low 32 bits == wave LDS offset
// (flat LDS aperture maps addr[31:0] -> LDS), so truncation gives VDST addr.
// ---------------------------------------------------------------------------
__device__ __forceinline__ uint32_t lds_off(const void* p) {
    return (uint32_t)(uintptr_t)p;
}

__device__ __forceinline__ void async_load_b32(uint32_t lds, const void* gaddr) {
    asm volatile("global_load_async_to_lds_b32 %0, %1, off"
                 :: "v"(lds), "v"((uint64_t)(uintptr_t)gaddr)
                 : "memory");
}

__device__ __forceinline__ void wait_async0() {
    asm volatile("s_wait_asynccnt 0" ::: "memory");
}

// ---------------------------------------------------------------------------
// Kernel 0: sel[b, ap] = -1  (no user assigned)
// ---------------------------------------------------------------------------
__global__ void k_init_sel(int* __restrict__ sel) {
    int i = blockIdx.x * blockDim.x + threadIdx.x;
    if (i < BATCH * NAP) sel[i] = -1;
}

// ---------------------------------------------------------------------------
// Kernel 1: per-user nearest-AP argmin + atomicMax selection.
// Block = 256 threads, grid = (NUD/256, BATCH). APs staged in LDS via async DMA.
// ---------------------------------------------------------------------------
__global__ void __launch_bounds__(256)
k_argmin(const float* __restrict__ Xap,
         const float* __restrict__ Xuser,
         int* __restrict__ sel) {
    __shared__ float sx[NAP];
    __shared__ float sy[NAP];
    __shared__ float sz[NAP];

    const int t = threadIdx.x;          // 0..255 : AP id for staging, user lane
    const int b = blockIdx.y;           // 0..31
    const int chunk = blockIdx.x;       // 0..9

    // Stage all 256 AP coordinates for batch b into LDS (async DMA, ASYNCcnt).
    {
        const float* ap = Xap + ((size_t)b * NAP + t) * 3;
        async_load_b32(lds_off(&sx[t]), ap + 0);
        async_load_b32(lds_off(&sy[t]), ap + 1);
        async_load_b32(lds_off(&sz[t]), ap + 2);
    }
    wait_async0();
    __syncthreads();

    const int j = chunk * 256 + t;      // user id 0..2559
    const float* us = Xuser + ((size_t)b * NUD + j) * 3;
    const float ux = us[0], uy = us[1], uz = us[2];

    float best = 3.4e38f;
    int   bi   = 0;
    #pragma unroll 8
    for (int i = 0; i < NAP; ++i) {
        float dx = fabsf(sx[i] - ux); dx = fminf(dx, EXW - dx);
        float dy = fabsf(sy[i] - uy); dy = fminf(dy, EXW - dy);
        float dz = sz[i] - uz;
        float d2 = fmaf(dz, dz, fmaf(dy, dy, dx * dx));
        if (d2 < best) { best = d2; bi = i; }   // ascending + strict < == first-min
    }
    // softmax(100*mask)+round picks the LARGEST user index assigned to AP bi.
    atomicMax(&sel[b * NAP + bi], j);
}

// ---------------------------------------------------------------------------
// Kernel 2: g_linear[b,a,i] = 1e-4.6 * d(AP a, user sel[b,i])^-3.8 ; diag -> 1/x.
// Block = 256 threads (i), grid = (16 a-tiles, BATCH). Selected-user coords
// async-GATHERED into LDS once per block, reused across 16 APs.
// ---------------------------------------------------------------------------
__global__ void __launch_bounds__(256)
k_output(const float* __restrict__ Xap,
         const float* __restrict__ Xuser,
         const int* __restrict__ sel,
         float* __restrict__ glin_out,
         float* __restrict__ pow_out) {
    __shared__ float ux[NAP];
    __shared__ float uy[NAP];
    __shared__ float uz[NAP];
    __shared__ int   sj[NAP];

    const int t    = threadIdx.x;       // i (column / selected-user slot)
    const int tile = blockIdx.x;        // 0..15 -> APs [tile*16, tile*16+16)
    const int b    = blockIdx.y;        // 0..31

    const int j  = sel[b * NAP + t];
    sj[t] = j;
    const int jj = (j < 0) ? 0 : j;     // keep all lanes active for the DMA
    const float* us = Xuser + ((size_t)b * NUD + jj) * 3;
    async_load_b32(lds_off(&ux[t]), us + 0);
    async_load_b32(lds_off(&uy[t]), us + 1);
    async_load_b32(lds_off(&uz[t]), us + 2);
    wait_async0();
    __syncthreads();

    const float C = -15.2808692f;       // -4.6 * log2(10)
    const float vx = ux[t], vy = uy[t], vz = uz[t];
    const bool  none = (sj[t] < 0);

    #pragma unroll
    for (int aa = 0; aa < 16; ++aa) {
        const int a = tile * 16 + aa;
        const float* ap = Xap + ((size_t)b * NAP + a) * 3;   // uniform -> s_load
        const float ax = ap[0], ay = ap[1], az = ap[2];

        float dx = fabsf(ax - vx); dx = fminf(dx, EXW - dx);
        float dy = fabsf(ay - vy); dy = fminf(dy, EXW - dy);
        float dz = az - vz;
        float d  = sqrtf(fmaf(dz, dz, fmaf(dy, dy, dx * dx)));
        if (none) d = 0.0f;             // -> log2 = -inf -> g_linear = +inf (ref)

        const float gl = exp2f(fmaf(-3.8f, __log2f(d), C));
        glin_out[((size_t)b * NAP + a) * NAP + t] = gl;
        if (t == a) pow_out[b * NAP + a] = 1.0f / gl;   // 1/inf = 0, matches ref
    }
}

// ---------------------------------------------------------------------------
extern "C" void kernel_launch(void* const* d_in, const int* in_sizes, int n_in,
                              void* d_out, int out_size, void* d_ws, size_t ws_size,
                              hipStream_t stream) {
    (void)in_sizes; (void)n_in; (void)out_size; (void)ws_size;

    const float* Xap   = (const float*)d_in[0];   // [64, 256, 3]
    const float* Xuser = (const float*)d_in[1];   // [64, 2560, 3]
    // d_in[2] = batch_num (always 32 per reference constants)

    int*   sel  = (int*)d_ws;                             // [32*256]
    float* glin = (float*)d_out;                          // [32,256,256]
    float* powp = glin + (size_t)BATCH * NAP * NAP;       // [32,256]

    k_init_sel<<<dim3((BATCH * NAP + 255) / 256), dim3(256), 0, stream>>>(sel);
    k_argmin  <<<dim3(NUD / 256, BATCH), dim3(256), 0, stream>>>(Xap, Xuser, sel);
    k_output  <<<dim3(NAP / 16, BATCH), dim3(256), 0, stream>>>(Xap, Xuser, sel,
                                                                glin, powp);
}